// GCN_41120016892386
// MI455X (gfx1250) — compile-verified
//
#include <hip/hip_runtime.h>
#include <math.h>

typedef __attribute__((ext_vector_type(2))) float v2f;
typedef __attribute__((ext_vector_type(8))) float v8f;

#define N_NODES 100000
#define N_EDGES 1600000
#define DIM 64

// ---------------- utility kernels ----------------

__global__ void zero_f32(float* __restrict__ p, int n) {
    int i = blockIdx.x * blockDim.x + threadIdx.x;
    if (i < n) p[i] = 0.0f;
}

__global__ void degree_kernel(float* __restrict__ deg, const int* __restrict__ col, int e) {
    int i = blockIdx.x * blockDim.x + threadIdx.x;
    if (i < e) atomicAdd(&deg[col[i]], 1.0f);
}

__global__ void dinv_kernel(float* __restrict__ deg, int n) {
    int i = blockIdx.x * blockDim.x + threadIdx.x;
    if (i < n) {
        float d = deg[i];
        deg[i] = (d > 0.0f) ? rsqrtf(d) : 0.0f;
    }
}

__global__ void norm_kernel(float* __restrict__ nrm, const float* __restrict__ dinv,
                            const int* __restrict__ row, const int* __restrict__ col, int e) {
    int i = blockIdx.x * blockDim.x + threadIdx.x;
    if (i < e) nrm[i] = dinv[row[i]] * dinv[col[i]];
}

// One thread per (edge, feature): gather h[row]*nrm, atomic scatter-add to col.
// Working set is L2-resident (25.6MB per feature buffer, 192MB L2).
__global__ void propagate_kernel(const float* __restrict__ src, float* __restrict__ dst,
                                 const int* __restrict__ row, const int* __restrict__ col,
                                 const float* __restrict__ nrm, int total) {
    int idx = blockIdx.x * blockDim.x + threadIdx.x;
    if (idx >= total) return;
    int e = idx >> 6;      // edge id
    int d = idx & 63;      // feature id
    float v = src[row[e] * DIM + d] * nrm[e];
    atomicAdd(&dst[col[e] * DIM + d], v);
}

// ---------------- WMMA f32 GEMM:  ACC (+)= H[N x 64] @ W[64 x 64] ----------------
// One wave computes a 16x16 output tile via 16 chained V_WMMA_F32_16X16X4_F32.
// A-fragment (16x4 f32, 2 VGPR): lane%16 = M row, lane<16 -> K={kk,kk+1}, lane>=16 -> K={kk+2,kk+3}
// B-fragment (4x16 f32, 2 VGPR): lane%16 = N col, lane<16 -> K rows {kk,kk+1}, else {kk+2,kk+3}
// C/D (16x16 f32, 8 VGPR): VGPR r, lanes 0-15 -> M=r, lanes 16-31 -> M=r+8; N = lane%16
__global__ void __launch_bounds__(256)
gemm_wmma_f32(const float* __restrict__ H, const float* __restrict__ W,
              float* __restrict__ ACC, int nTiles, int beta) {
    int wave = (blockIdx.x * blockDim.x + threadIdx.x) >> 5;
    if (wave >= nTiles) return;                    // wave-uniform: EXEC stays all-1s
    int lane   = threadIdx.x & 31;
    int m0     = (wave >> 2) << 4;                 // 4 col-tiles per 16-row strip
    int n0     = (wave & 3) << 4;
    int l16    = lane & 15;
    int hiHalf = lane >> 4;                        // 0 or 1

    v8f c = {};
    if (beta) {
#pragma unroll
        for (int r = 0; r < 8; ++r)
            c[r] = ACC[(m0 + r + hiHalf * 8) * DIM + n0 + l16];
    }

    const float* hrow = H + (m0 + l16) * DIM;
#pragma unroll
    for (int kk = 0; kk < DIM; kk += 4) {
        int kb = kk + hiHalf * 2;                  // even -> 8B aligned
        v2f a = *(const v2f*)(hrow + kb);          // A[m, kb], A[m, kb+1]
        v2f b;
        b.x = W[kb * DIM + n0 + l16];              // B[kb,   n]
        b.y = W[(kb + 1) * DIM + n0 + l16];        // B[kb+1, n]
        c = __builtin_amdgcn_wmma_f32_16x16x4_f32(
            /*neg_a=*/false, a, /*neg_b=*/false, b,
            /*c_mod=*/(short)0, c, /*reuse_a=*/false, /*reuse_b=*/false);
    }

#pragma unroll
    for (int r = 0; r < 8; ++r)
        ACC[(m0 + r + hiHalf * 8) * DIM + n0 + l16] = c[r];
}

// ---------------- epilogues ----------------

__global__ void bias_tanh_kernel(const float* __restrict__ acc, const float* __restrict__ b,
                                 float* __restrict__ out, int total) {
    int i = blockIdx.x * blockDim.x + threadIdx.x;
    if (i < total) out[i] = tanhf(acc[i] + b[i & 63]);
}

__global__ void bias_add_kernel(const float* __restrict__ acc, const float* __restrict__ b,
                                float* __restrict__ out, int total) {
    int i = blockIdx.x * blockDim.x + threadIdx.x;
    if (i < total) out[i] = acc[i] + b[i & 63];
}

// ---------------- host orchestration ----------------

extern "C" void kernel_launch(void* const* d_in, const int* in_sizes, int n_in,
                              void* d_out, int out_size, void* d_ws, size_t ws_size,
                              hipStream_t stream) {
    const float* x  = (const float*)d_in[0];
    const int*   ei = (const int*)d_in[1];      // [2, E] int32
    const float* w1 = (const float*)d_in[2];    // [11, 64, 64]
    const float* b1 = (const float*)d_in[3];
    const float* w2 = (const float*)d_in[4];    // [4, 64, 64]
    const float* b2 = (const float*)d_in[5];
    const float* wc = (const float*)d_in[6];    // [64, 64]
    const float* bc = (const float*)d_in[7];
    float* out = (float*)d_out;

    const int N = N_NODES, E = N_EDGES;
    const int ND = N * DIM;
    const int* row = ei;
    const int* col = ei + E;

    // carve scratch (all offsets multiples of 256B)
    float* dinv = (float*)d_ws;                        // N   (padded)
    float* nrm  = dinv + ((N + 63) & ~63);             // E
    float* HA   = nrm + E;                             // N*D ping
    float* HB   = HA + (size_t)ND;                     // N*D pong
    float* ACC  = HB + (size_t)ND;                     // N*D accumulator
    float* XL   = ACC + (size_t)ND;                    // N*D layer output

    const int T = 256;
    const int nTiles = (N / 16) * 4;                   // 25000 waves
    const int gemmBlocks = nTiles / 8;                 // 8 waves per block, exact
    const int edBlocks = (E * DIM + T - 1) / T;

    // ---- normalization ----
    zero_f32<<<(N + T - 1) / T, T, 0, stream>>>(dinv, N);
    degree_kernel<<<(E + T - 1) / T, T, 0, stream>>>(dinv, col, E);
    dinv_kernel<<<(N + T - 1) / T, T, 0, stream>>>(dinv, N);
    norm_kernel<<<(E + T - 1) / T, T, 0, stream>>>(nrm, dinv, row, col, E);

    // ---- TAG layer 1: K=10 ----
    gemm_wmma_f32<<<gemmBlocks, T, 0, stream>>>(x, w1, ACC, nTiles, 0);
    const float* cur = x;
    for (int k = 1; k <= 10; ++k) {
        float* tgt = (k & 1) ? HA : HB;
        zero_f32<<<(ND + T - 1) / T, T, 0, stream>>>(tgt, ND);
        propagate_kernel<<<edBlocks, T, 0, stream>>>(cur, tgt, row, col, nrm, E * DIM);
        gemm_wmma_f32<<<gemmBlocks, T, 0, stream>>>(tgt, w1 + (size_t)k * DIM * DIM, ACC, nTiles, 1);
        cur = tgt;
    }
    bias_tanh_kernel<<<(ND + T - 1) / T, T, 0, stream>>>(ACC, b1, XL, ND);

    // ---- TAG layer 2: K=3 ----
    gemm_wmma_f32<<<gemmBlocks, T, 0, stream>>>(XL, w2, ACC, nTiles, 0);
    cur = XL;
    for (int k = 1; k <= 3; ++k) {
        float* tgt = (k & 1) ? HA : HB;
        zero_f32<<<(ND + T - 1) / T, T, 0, stream>>>(tgt, ND);
        propagate_kernel<<<edBlocks, T, 0, stream>>>(cur, tgt, row, col, nrm, E * DIM);
        gemm_wmma_f32<<<gemmBlocks, T, 0, stream>>>(tgt, w2 + (size_t)k * DIM * DIM, ACC, nTiles, 1);
        cur = tgt;
    }
    bias_tanh_kernel<<<(ND + T - 1) / T, T, 0, stream>>>(ACC, b2, XL, ND);

    // ---- final linear ----
    gemm_wmma_f32<<<gemmBlocks, T, 0, stream>>>(XL, wc, ACC, nTiles, 0);
    bias_add_kernel<<<(ND + T - 1) / T, T, 0, stream>>>(ACC, bc, out, ND);
}